// Min_interval_66640712564825
// MI455X (gfx1250) — compile-verified
//
#include <hip/hip_runtime.h>

#define BATCH   65536
#define NCOLS   16
#define OUTC    696      // 16 + C(16,2) + C(16,3) = 16 + 120 + 560
#define BLOCK   256
#define CH      24       // columns per LDS-staged chunk; 696 = 24 * 29
#define CHP     (CH / 2) // column pairs per chunk (for b64 flush)
#define PITCH   26       // even -> 8B-aligned pairs; 26·Δt≢0 (mod 64) for Δt<32 -> conflict-free

// r <- lexmin(r, c) by (cur, beta). Exact ties imply identical (l,u), so
// tie direction cannot change output values (matches reference semantics).
__device__ __forceinline__ void fold_min(float& rl, float& ru, float& rc, float& rb,
                                         float cl, float cu, float cc2, float cb) {
  const bool take = (cc2 < rc) || ((cc2 == rc) && (cb < rb));
  rl = take ? cl : rl;
  ru = take ? cu : ru;
  rc = take ? cc2 : rc;
  rb = take ? cb : rb;
}

// Stage result column c into LDS; every CH columns, flush the chunk with
// lane-contiguous b64 global stores. c is a compile-time constant at every
// expansion site after full unroll, so cc_/c0_ fold to immediates. The flush
// copy loop is deliberately kept rolled (#pragma unroll 1) to bound static
// code size: the 696-column compute stream must be straight-line, the 29
// identical copy loops need not be.
#define EMIT(RL, RU)                                                          \
  {                                                                           \
    const int cc_ = c % CH;                                                   \
    sl[tid * PITCH + cc_] = (RL);                                             \
    su[tid * PITCH + cc_] = (RU);                                             \
    if (cc_ == CH - 1) {                                                      \
      __syncthreads();                                                        \
      const int c0_ = c - (CH - 1);                                           \
      _Pragma("unroll 1")                                                     \
      for (int it = 0; it < CHP; ++it) {                                      \
        const int f   = it * BLOCK + tid;   /* 0 .. 256*CHP-1 */              \
        const int r   = f / CHP;            /* row within block   */          \
        const int col = 2 * (f - r * CHP);  /* even column offset */          \
        const size_t g = (size_t)(rbase + r) * OUTC + (size_t)(c0_ + col);    \
        const float2 vl = *reinterpret_cast<const float2*>(&sl[r * PITCH + col]); \
        const float2 vu = *reinterpret_cast<const float2*>(&su[r * PITCH + col]); \
        *reinterpret_cast<float2*>(&outl[g]) = vl;                            \
        *reinterpret_cast<float2*>(&outu[g]) = vu;                            \
      }                                                                       \
      __syncthreads();                                                        \
    }                                                                         \
    ++c;                                                                      \
  }

__global__ __launch_bounds__(BLOCK) void minintv_kernel(
    const float* __restrict__ xl, const float* __restrict__ xu,
    float* __restrict__ out) {
  __shared__ __align__(16) float sl[BLOCK * PITCH];
  __shared__ __align__(16) float su[BLOCK * PITCH];
  const int tid   = threadIdx.x;
  const int rbase = blockIdx.x * BLOCK;
  const int row   = rbase + tid;

  // gfx1250: global_prefetch_b8 for the input rows.
  __builtin_prefetch(xl + (size_t)row * NCOLS, 0, 3);
  __builtin_prefetch(xu + (size_t)row * NCOLS, 0, 3);

  // Per-row state held entirely in VGPRs (64 floats).
  float lv[NCOLS], uv[NCOLS], cv[NCOLS], bv[NCOLS];
  const float4* pl = reinterpret_cast<const float4*>(xl) + (size_t)row * 4;
  const float4* pu = reinterpret_cast<const float4*>(xu) + (size_t)row * 4;
#pragma unroll
  for (int q = 0; q < 4; ++q) {
    const float4 a = pl[q];
    const float4 b = pu[q];
    lv[4 * q + 0] = a.x; lv[4 * q + 1] = a.y; lv[4 * q + 2] = a.z; lv[4 * q + 3] = a.w;
    uv[4 * q + 0] = b.x; uv[4 * q + 1] = b.y; uv[4 * q + 2] = b.z; uv[4 * q + 3] = b.w;
  }
  {
    // Reference computes cur/beta with separate mul+add roundings; an FMA
    // contraction here can flip the beta tie-break on exact cur ties, which
    // swaps a whole interval. Round-2 asm showed v_fmac contraction, so
    // forbid it explicitly for this block.
#pragma clang fp contract(off)
#pragma unroll
    for (int i = 0; i < NCOLS; ++i) {
      // cur: 0.5*(l+u) == fl(0.5l)+fl(0.5u) bit-exactly (halving is exact).
      cv[i] = 0.5f * (lv[i] + uv[i]);
      bv[i] = 0.2f * lv[i] + 0.8f * uv[i];
    }
  }

  float* __restrict__ outl = out;
  float* __restrict__ outu = out + (size_t)BATCH * (size_t)OUTC;

  // Enumerate subsets in ascending-bitmask order (== reference output order):
  // for highest bit h: {h}; then {h2,h} for h2<h; then {h3,h2,h} for h3<h2.
  int c = 0;
#pragma unroll
  for (int h = 0; h < NCOLS; ++h) {
    EMIT(lv[h], uv[h]);                       // singleton {h}
#pragma unroll
    for (int h2 = 0; h2 < h; ++h2) {
      // pair {h2,h}: lexmin(col_h2, col_h)
      float rl = lv[h2], ru = uv[h2], rc = cv[h2], rb = bv[h2];
      fold_min(rl, ru, rc, rb, lv[h], uv[h], cv[h], bv[h]);
      EMIT(rl, ru);
#pragma unroll
      for (int h3 = 0; h3 < h2; ++h3) {
        // triple {h3,h2,h}: lexmin(pair{h2,h}, col_h3) — one fold, reusing rl..rb
        float tl = rl, tu = ru, tc = rc, tb = rb;
        fold_min(tl, tu, tc, tb, lv[h3], uv[h3], cv[h3], bv[h3]);
        EMIT(tl, tu);
      }
    }
  }
}

extern "C" void kernel_launch(void* const* d_in, const int* in_sizes, int n_in,
                              void* d_out, int out_size, void* d_ws, size_t ws_size,
                              hipStream_t stream) {
  const float* xl = (const float*)d_in[0];
  const float* xu = (const float*)d_in[1];
  float* out = (float*)d_out;
  (void)in_sizes; (void)n_in; (void)out_size; (void)d_ws; (void)ws_size;
  minintv_kernel<<<dim3(BATCH / BLOCK), dim3(BLOCK), 0, stream>>>(xl, xu, out);
}